// AttentionCrossSpatial_14336600834768
// MI455X (gfx1250) — compile-verified
//
#include <hip/hip_runtime.h>
#include <hip/hip_bf16.h>

typedef float v2f __attribute__((ext_vector_type(2)));
typedef float v4f __attribute__((ext_vector_type(4)));
typedef float v8f __attribute__((ext_vector_type(8)));

// ---------------------------------------------------------------------------
// CDNA5 f32 WMMA: D(16x16,f32) = A(16x4,f32) * B(4x16,f32) + C
// A layout (ISA 7.12.2): lanes 0-15 rows M=0..15 {K=0,K=1}; lanes 16-31 {K=2,K=3}
// B mirrors (rows K striped across half-waves); C/D: VGPR r -> M=r / M=r+8.
// ---------------------------------------------------------------------------
__device__ __forceinline__ v8f wmma4(v2f a, v2f b, v8f c) {
  return __builtin_amdgcn_wmma_f32_16x16x4_f32(false, a, false, b, (short)0, c,
                                               false, false);
}

#define ZACC {0.f, 0.f, 0.f, 0.f, 0.f, 0.f, 0.f, 0.f}

// ===========================================================================
// Row-major GEMM: C[M,N] = W[M,KK] @ X[KK,NN], M multiple of 16.
// KK/NN compile-time -> no remainder tail, constant strides.
// Block = 4 waves; each wave owns 16x64 (one A-frag feeds 4 wmmas).
// grid = (NN/256, M/16).  Used for conv1x1 (192x64) and final proj (64x64).
// ===========================================================================
template <int KK, int NN>
__global__ __launch_bounds__(128) void gemm_wn(const float* __restrict__ W,
                                               const float* __restrict__ X,
                                               float* __restrict__ C, int M) {
  const int lane = threadIdx.x & 31, wave = threadIdx.x >> 5;
  const int la = lane & 15;
  const int hi = lane >> 4;  // 0/1
  const int m0 = blockIdx.y * 16;
  const int n0 = (blockIdx.x * 4 + wave) * 64;
  const float* ap = W + (m0 + la) * KK + hi * 2;
  const float* bp = X + hi * 2 * NN + n0 + la;
  v8f a0 = ZACC, a1 = ZACC, a2 = ZACC, a3 = ZACC;
#pragma unroll 2
  for (int k = 0; k < KK; k += 4) {
    const v2f a = *(const v2f*)ap;
    v2f b0, b1, b2, b3;
    b0.x = bp[0];  b0.y = bp[NN];
    b1.x = bp[16]; b1.y = bp[NN + 16];
    b2.x = bp[32]; b2.y = bp[NN + 32];
    b3.x = bp[48]; b3.y = bp[NN + 48];
    a0 = wmma4(a, b0, a0);
    a1 = wmma4(a, b1, a1);
    a2 = wmma4(a, b2, a2);
    a3 = wmma4(a, b3, a3);
    ap += 4;
    bp += 4 * NN;
  }
  float* cp = C + (m0 + hi * 8) * NN + n0 + la;
#pragma unroll
  for (int r = 0; r < 8; ++r) {
    cp[r * NN] = a0[r];
    cp[r * NN + 16] = a1[r];
    cp[r * NN + 32] = a2[r];
    cp[r * NN + 48] = a3[r];
  }
}

// ===========================================================================
// Batched (z=channel) C[z] = scale * A[z] @ B[z]^T, 256x256, K=256.
// All fragment loads are contiguous v2f (global_load_b64).
// grid = (1, 16, 64); wave owns 16x64 (4 accumulators).
// ===========================================================================
__global__ __launch_bounds__(128) void gemm_abt_batched(
    const float* __restrict__ A, const float* __restrict__ B,
    float* __restrict__ C, float scale) {
  const int z = blockIdx.z;
  const int lane = threadIdx.x & 31, wave = threadIdx.x >> 5;
  const int la = lane & 15;
  const int hi = lane >> 4;
  const int m0 = blockIdx.y * 16;
  const int n0 = wave * 64;
  const float* ap = A + z * 65536 + (m0 + la) * 256 + hi * 2;
  const float* bp = B + z * 65536 + (n0 + la) * 256 + hi * 2;
  v8f a0 = ZACC, a1 = ZACC, a2 = ZACC, a3 = ZACC;
#pragma unroll 2
  for (int k = 0; k < 256; k += 4) {
    const v2f a = *(const v2f*)ap;
    const v2f b0 = *(const v2f*)(bp);
    const v2f b1 = *(const v2f*)(bp + 16 * 256);
    const v2f b2 = *(const v2f*)(bp + 32 * 256);
    const v2f b3 = *(const v2f*)(bp + 48 * 256);
    a0 = wmma4(a, b0, a0);
    a1 = wmma4(a, b1, a1);
    a2 = wmma4(a, b2, a2);
    a3 = wmma4(a, b3, a3);
    ap += 4;
    bp += 4;
  }
  float* cp = C + z * 65536 + (m0 + hi * 8) * 256 + n0 + la;
#pragma unroll
  for (int r = 0; r < 8; ++r) {
    cp[r * 256] = a0[r] * scale;
    cp[r * 256 + 16] = a1[r] * scale;
    cp[r * 256 + 32] = a2[r] * scale;
    cp[r * 256 + 48] = a3[r] * scale;
  }
}

// ===========================================================================
// Batched row-major C[z] = A[z] @ B[z], 256x256x256.  Used for (proj*mask)@V.
// grid = (1, 16, 64).
// ===========================================================================
__global__ __launch_bounds__(128) void gemm_ab_batched(
    const float* __restrict__ A, const float* __restrict__ B,
    float* __restrict__ C) {
  const int z = blockIdx.z;
  const int lane = threadIdx.x & 31, wave = threadIdx.x >> 5;
  const int la = lane & 15;
  const int hi = lane >> 4;
  const int m0 = blockIdx.y * 16;
  const int n0 = wave * 64;
  const float* ap = A + z * 65536 + (m0 + la) * 256 + hi * 2;
  const float* bp = B + z * 65536 + hi * 2 * 256 + n0 + la;
  v8f a0 = ZACC, a1 = ZACC, a2 = ZACC, a3 = ZACC;
#pragma unroll 2
  for (int k = 0; k < 256; k += 4) {
    const v2f a = *(const v2f*)ap;
    v2f b0, b1, b2, b3;
    b0.x = bp[0];  b0.y = bp[256];
    b1.x = bp[16]; b1.y = bp[256 + 16];
    b2.x = bp[32]; b2.y = bp[256 + 32];
    b3.x = bp[48]; b3.y = bp[256 + 48];
    a0 = wmma4(a, b0, a0);
    a1 = wmma4(a, b1, a1);
    a2 = wmma4(a, b2, a2);
    a3 = wmma4(a, b3, a3);
    ap += 4;
    bp += 4 * 256;
  }
  float* cp = C + z * 65536 + (m0 + hi * 8) * 256 + n0 + la;
#pragma unroll
  for (int r = 0; r < 8; ++r) {
    cp[r * 256] = a0[r];
    cp[r * 256 + 16] = a1[r];
    cp[r * 256 + 32] = a2[r];
    cp[r * 256 + 48] = a3[r];
  }
}

// ===========================================================================
// FRMap: Yt[z](100x256) = WfrP(112x100, zero-padded rows) @ Ut[z](100x256).
// grid = (1, 7, 64); unguarded A loads thanks to the padded weight copy.
// ===========================================================================
__global__ __launch_bounds__(128) void frmap_gemm(const float* __restrict__ WfrP,
                                                  const float* __restrict__ Ut,
                                                  float* __restrict__ Yt) {
  const int z = blockIdx.z;
  const int lane = threadIdx.x & 31, wave = threadIdx.x >> 5;
  const int la = lane & 15;
  const int hi = lane >> 4;
  const int m0 = blockIdx.y * 16;
  const int n0 = wave * 64;
  const float* ap = WfrP + (m0 + la) * 100 + hi * 2;
  const float* bp = Ut + z * 25600 + hi * 2 * 256 + n0 + la;
  v8f a0 = ZACC, a1 = ZACC, a2 = ZACC, a3 = ZACC;
#pragma unroll 1
  for (int k = 0; k < 100; k += 4) {  // 25 steps, K = 0..99 exactly
    const v2f a = *(const v2f*)ap;
    v2f b0, b1, b2, b3;
    b0.x = bp[0];  b0.y = bp[256];
    b1.x = bp[16]; b1.y = bp[256 + 16];
    b2.x = bp[32]; b2.y = bp[256 + 32];
    b3.x = bp[48]; b3.y = bp[256 + 48];
    a0 = wmma4(a, b0, a0);
    a1 = wmma4(a, b1, a1);
    a2 = wmma4(a, b2, a2);
    a3 = wmma4(a, b3, a3);
    ap += 4;
    bp += 4 * 256;
  }
  float* cp = Yt + z * 25600 + (m0 + hi * 8) * 256 + n0 + la;
#pragma unroll
  for (int r = 0; r < 8; ++r) {
    const int row = m0 + hi * 8 + r;
    if (row < 100) {
      cp[r * 256] = a0[r];
      cp[r * 256 + 16] = a1[r];
      cp[r * 256 + 32] = a2[r];
      cp[r * 256 + 48] = a3[r];
    }
  }
}

// ===========================================================================
// Projmap with mask fused: P[z][i][j] = s(i,j) * sum_o Yt[o][i] Yt[o][j],
// s = +1 diag / -1 off-diag (mask = 2I-1).  grid = (1, 16, 64).
// ===========================================================================
__global__ __launch_bounds__(128) void proj_masked(const float* __restrict__ Yt,
                                                   float* __restrict__ P) {
  const int z = blockIdx.z;
  const int lane = threadIdx.x & 31, wave = threadIdx.x >> 5;
  const int la = lane & 15;
  const int hi = lane >> 4;
  const int m0 = blockIdx.y * 16;
  const int n0 = wave * 64;
  const float* ap = Yt + z * 25600 + hi * 2 * 256 + m0 + la;
  const float* bp = Yt + z * 25600 + hi * 2 * 256 + n0 + la;
  v8f a0 = ZACC, a1 = ZACC, a2 = ZACC, a3 = ZACC;
#pragma unroll 1
  for (int k = 0; k < 100; k += 4) {
    v2f a, b0, b1, b2, b3;
    a.x = ap[0];   a.y = ap[256];
    b0.x = bp[0];  b0.y = bp[256];
    b1.x = bp[16]; b1.y = bp[256 + 16];
    b2.x = bp[32]; b2.y = bp[256 + 32];
    b3.x = bp[48]; b3.y = bp[256 + 48];
    a0 = wmma4(a, b0, a0);
    a1 = wmma4(a, b1, a1);
    a2 = wmma4(a, b2, a2);
    a3 = wmma4(a, b3, a3);
    ap += 4 * 256;
    bp += 4 * 256;
  }
  float* cp = P + z * 65536 + (m0 + hi * 8) * 256 + n0 + la;
  const int i0 = m0 + hi * 8;
#pragma unroll
  for (int r = 0; r < 8; ++r) {
    const int i = i0 + r, j0 = n0 + la;
    cp[r * 256]      = (i == j0)      ? a0[r] : -a0[r];
    cp[r * 256 + 16] = (i == j0 + 16) ? a1[r] : -a1[r];
    cp[r * 256 + 32] = (i == j0 + 32) ? a2[r] : -a2[r];
    cp[r * 256 + 48] = (i == j0 + 48) ? a3[r] : -a3[r];
  }
}

// ===========================================================================
// Zero-pad w_fr (100x100) into WfrP (112x100) so FRMap needs no A guards.
// ===========================================================================
__global__ void pad_wfr(const float* __restrict__ Wfr, float* __restrict__ P) {
  const int i = blockIdx.x * 256 + threadIdx.x;
  if (i >= 112 * 100) return;
  P[i] = (i < 10000) ? Wfr[i] : 0.f;
}

// ===========================================================================
// Depthwise 3x3 SAME conv fused with patchify + q/k/v split.
// ===========================================================================
__global__ void dwconv_patchify(const float* __restrict__ X,
                                const float* __restrict__ Wd,
                                float* __restrict__ Q, float* __restrict__ Ko,
                                float* __restrict__ V) {
  const int idx = blockIdx.x * 256 + threadIdx.x;  // < 192*65536
  const int ch = idx >> 16;
  const int hw = idx & 65535;
  const int h = hw >> 8, w = hw & 255;
  const float* xin = X + ch * 65536;
  const float* wd = Wd + ch * 9;
  float acc = 0.f;
#pragma unroll
  for (int dy = 0; dy < 3; ++dy) {
    const int hy = h + dy - 1;
    if (hy < 0 || hy > 255) continue;
#pragma unroll
    for (int dx = 0; dx < 3; ++dx) {
      const int wx = w + dx - 1;
      if (wx < 0 || wx > 255) continue;
      acc += xin[hy * 256 + wx] * wd[dy * 3 + dx];
    }
  }
  const int grp = ch >> 6;  // 0=q 1=k 2=v
  const int c = ch & 63;
  const int n = (h >> 4) * 16 + (w >> 4);
  const int d = (h & 15) * 16 + (w & 15);
  float* dst = (grp == 0) ? Q : (grp == 1 ? Ko : V);
  dst[c * 65536 + n * 256 + d] = acc;
}

// ===========================================================================
// L2-normalize each 256-element row (one block per row).
// ===========================================================================
__global__ void l2norm_rows(float* __restrict__ X) {
  float* r = X + blockIdx.x * 256;
  __shared__ float red[256];
  const float v = r[threadIdx.x];
  red[threadIdx.x] = v * v;
  __syncthreads();
  for (int s = 128; s > 0; s >>= 1) {
    if (threadIdx.x < s) red[threadIdx.x] += red[threadIdx.x + s];
    __syncthreads();
  }
  const float nrm = sqrtf(red[0]);
  r[threadIdx.x] = v / fmaxf(nrm, 1e-12f);
}

// ===========================================================================
// Row softmax (temperature indexed by cph = c&7) fused with row-mean centering.
// ===========================================================================
__global__ void softmax_center(float* __restrict__ S, const float* __restrict__ temp) {
  const int row = blockIdx.x;
  const float tv = temp[(row >> 8) & 7];
  float* r = S + row * 256;
  __shared__ float red[256];
  const float v = r[threadIdx.x] * tv;
  red[threadIdx.x] = v;
  __syncthreads();
  for (int s = 128; s > 0; s >>= 1) {
    if (threadIdx.x < s) red[threadIdx.x] = fmaxf(red[threadIdx.x], red[threadIdx.x + s]);
    __syncthreads();
  }
  const float mx = red[0];
  __syncthreads();
  const float e = expf(v - mx);
  red[threadIdx.x] = e;
  __syncthreads();
  for (int s = 128; s > 0; s >>= 1) {
    if (threadIdx.x < s) red[threadIdx.x] += red[threadIdx.x + s];
    __syncthreads();
  }
  const float p = e / red[0];
  __syncthreads();
  red[threadIdx.x] = p;
  __syncthreads();
  for (int s = 128; s > 0; s >>= 1) {
    if (threadIdx.x < s) red[threadIdx.x] += red[threadIdx.x + s];
    __syncthreads();
  }
  r[threadIdx.x] = p - red[0] * (1.0f / 256.0f);
}

// ===========================================================================
// Batched Jacobi eigensolver: 1024 threads (32 waves) per 256x256 SPD matrix.
// A in LDS, padded to 256x260 (1040B rows -> 16B-aligned, so row/eigenvector
// rotations use b128 LDS/global accesses; ~266KB of the 320KB WGP LDS).
// Chess-tournament ordering: 128 disjoint Givens rotations per step applied
// as J^T A (rows, float4), (.)J (cols, scalar strided), Vt <- J^T Vt (float4,
// L2-resident global).  Fuses trace-normalize + 1e-5 I + top-100 rank-select.
// ===========================================================================
#define JAC_SWEEPS 8
#define APITCH 260
#define JAC_SMEM_BYTES ((256 * APITCH + 1024 + 128 + 128) * 4 + (256 + 128 + 128) * 4)

__global__ __launch_bounds__(1024) void jacobi_topk(const float* __restrict__ Cov,
                                                    float* __restrict__ Vt,
                                                    float* __restrict__ Ut) {
  extern __shared__ float smem[];
  float* A = smem;                    // 256*260
  float* red = A + 256 * APITCH;      // 1024
  float* cs = red + 1024;             // 128
  float* sn = cs + 128;               // 128
  int* idx = (int*)(sn + 128);        // 256
  int* pr = idx + 256;                // 128 (reused as rank->row map)
  int* qr = pr + 128;                 // 128

  const int z = blockIdx.x;
  const int tid = threadIdx.x;
  const float* Cz = Cov + z * 65536;
  float* Vz = Vt + z * 65536;
  float* Uz = Ut + z * 25600;

  // Load (float4) + trace
  float trp = 0.f;
  for (int t = tid; t < 16384; t += 1024) {
    const int e = t * 4;
    const int rr = e >> 8, c0 = e & 255;
    const v4f v = *(const v4f*)(Cz + e);
    *(v4f*)(A + rr * APITCH + c0) = v;
    if (rr >= c0 && rr < c0 + 4) trp += v[rr - c0];
  }
  red[tid] = trp;
  __syncthreads();
  for (int s = 512; s > 0; s >>= 1) {
    if (tid < s) red[tid] += red[tid + s];
    __syncthreads();
  }
  const float inv_tr = 1.0f / red[0];
  __syncthreads();
  // Trace-normalize + eps*I (float4); Vt = I (float4)
  for (int t = tid; t < 16384; t += 1024) {
    const int e = t * 4;
    const int rr = e >> 8, c0 = e & 255;
    v4f v = *(v4f*)(A + rr * APITCH + c0);
    v4f vi;
#pragma unroll
    for (int u = 0; u < 4; ++u) {
      float x = v[u] * inv_tr;
      const bool dg = (rr == c0 + u);
      if (dg) x += 1e-5f;
      v[u] = x;
      vi[u] = dg ? 1.0f : 0.0f;
    }
    *(v4f*)(A + rr * APITCH + c0) = v;
    *(v4f*)(Vz + e) = vi;
  }
  if (tid < 256) idx[tid] = tid;
  __syncthreads();

  for (int sweep = 0; sweep < JAC_SWEEPS; ++sweep) {
#pragma unroll 1
    for (int step = 0; step < 255; ++step) {
      if (tid < 128) {
        int p = idx[tid], q = idx[255 - tid];
        if (p > q) { int t = p; p = q; q = t; }
        pr[tid] = p;
        qr[tid] = q;
        const float apq = A[p * APITCH + q];
        float c = 1.0f, s = 0.0f;
        if (fabsf(apq) > 1e-20f) {
          const float theta = (A[q * APITCH + q] - A[p * APITCH + p]) / (2.0f * apq);
          const float t = copysignf(1.0f, theta) /
                          (fabsf(theta) + sqrtf(1.0f + theta * theta));
          c = rsqrtf(1.0f + t * t);
          s = t * c;
        }
        cs[tid] = c;
        sn[tid] = s;
      }
      __syncthreads();
      // Rows: A <- J^T A  (float4, b128 LDS)
#pragma unroll 1
      for (int t = tid; t < 128 * 64; t += 1024) {
        const int k = t >> 6, j4 = (t & 63) * 4;
        const int p = pr[k], q = qr[k];
        const float c = cs[k], s = sn[k];
        const v4f ap = *(v4f*)(A + p * APITCH + j4);
        const v4f aq = *(v4f*)(A + q * APITCH + j4);
        *(v4f*)(A + p * APITCH + j4) = c * ap - s * aq;
        *(v4f*)(A + q * APITCH + j4) = s * ap + c * aq;
      }
      __syncthreads();
      // Cols: A <- A J  (scalar, strided)
#pragma unroll 1
      for (int t = tid; t < 128 * 256; t += 1024) {
        const int k = t >> 8, i = t & 255;
        const int p = pr[k], q = qr[k];
        const float c = cs[k], s = sn[k];
        const float ip = A[i * APITCH + p], iq = A[i * APITCH + q];
        A[i * APITCH + p] = c * ip - s * iq;
        A[i * APITCH + q] = s * ip + c * iq;
      }
      // Eigenvectors: Vt <- J^T Vt  (float4, b128 global, L2-resident)
#pragma unroll 1
      for (int t = tid; t < 128 * 64; t += 1024) {
        const int k = t >> 6, j4 = (t & 63) * 4;
        const int p = pr[k], q = qr[k];
        const float c = cs[k], s = sn[k];
        const v4f vp = *(const v4f*)(Vz + p * 256 + j4);
        const v4f vq = *(const v4f*)(Vz + q * 256 + j4);
        *(v4f*)(Vz + p * 256 + j4) = c * vp - s * vq;
        *(v4f*)(Vz + q * 256 + j4) = s * vp + c * vq;
      }
      __syncthreads();
      // Rotate tournament ring (position 0 fixed)
      int old = 0;
      if (tid < 256) old = idx[tid];
      __syncthreads();
      if (tid < 256) idx[(tid == 0) ? 0 : (tid == 255 ? 1 : tid + 1)] = old;
      __syncthreads();
    }
  }

  // Rank-select top-100 eigenvectors (descending eigenvalues).
  if (tid < 256) red[tid] = A[tid * APITCH + tid];
  __syncthreads();
  if (tid < 256) {
    const float ev = red[tid];
    int rank = 0;
    for (int j = 0; j < 256; ++j) {
      const float evj = red[j];
      if (evj > ev || (evj == ev && j < tid)) ++rank;
    }
    if (rank < 100) pr[rank] = tid;  // inverse map rank -> source row
  }
  __syncthreads();
  for (int t = tid; t < 6400; t += 1024) {
    const int r = t >> 6, c4 = (t & 63) * 4;
    *(v4f*)(Uz + r * 256 + c4) = *(const v4f*)(Vz + pr[r] * 256 + c4);
  }
}

// ===========================================================================
// Final shuffle (float4): 4 consecutive d' map to 4 consecutive w in one
// 16-block, so load and store are both aligned b128.
// ===========================================================================
__global__ void gather_shuffle4(const float* __restrict__ O, float* __restrict__ G) {
  const int t = blockIdx.x * 256 + threadIdx.x;  // < 64*65536/4
  const int e = t * 4;
  const int c = e >> 16;
  const int nd = e & 65535;
  const int n = nd >> 8, d0 = nd & 255;  // d0 multiple of 4
  const int h = (n >> 4) * 16 + (d0 >> 4);
  const int w0 = (n & 15) * 16 + (d0 & 15);
  *(v4f*)(G + e) = *(const v4f*)(O + c * 65536 + h * 256 + w0);
}

// ===========================================================================
extern "C" void kernel_launch(void* const* d_in, const int* in_sizes, int n_in,
                              void* d_out, int out_size, void* d_ws, size_t ws_size,
                              hipStream_t stream) {
  (void)in_sizes; (void)n_in; (void)out_size; (void)ws_size;
  const float* xir   = (const float*)d_in[0];
  const float* xvi   = (const float*)d_in[1];
  const float* wqkv  = (const float*)d_in[2];
  const float* wdw   = (const float*)d_in[3];
  const float* wproj = (const float*)d_in[4];
  const float* temp  = (const float*)d_in[5];
  const float* wfr   = (const float*)d_in[6];
  float* out = (float*)d_out;
  float* ws = (float*)d_ws;

  const size_t CH = 4194304;  // 64 * 256 * 256
  float* qkvpre = ws;              // 3*CH conv scratch, re-partitioned below
  float* attnXc = ws;              // CH
  float* covB   = ws + CH;         // CH
  float* projM  = ws + 2 * CH;     // CH
  float* Obuf   = attnXc;          // reuse (attn/xc dead after cov GEMM)
  float* Gbuf   = covB;            // reuse (cov dead after jacobi)
  float* Qb[2] = {ws + 3 * CH, ws + 6 * CH};
  float* Kb[2] = {ws + 4 * CH, ws + 7 * CH};
  float* Vb[2] = {ws + 5 * CH, ws + 8 * CH};
  float* Vt = ws + 9 * CH;                 // 64 x 256 x 256
  float* Ut = Vt + CH;                     // 64 x 100 x 256
  float* Yt = Ut + (size_t)64 * 25600;     // 64 x 100 x 256
  float* WfrP = Yt + (size_t)64 * 25600;   // 112 x 100 zero-padded

  pad_wfr<<<(112 * 100 + 255) / 256, 256, 0, stream>>>(wfr, WfrP);

  // ---- Preprocess both inputs: conv1x1 -> dwconv3x3 -> patchify -> l2norm
  for (int inp = 0; inp < 2; ++inp) {
    const float* x = inp ? xvi : xir;
    gemm_wn<64, 65536><<<dim3(65536 / 256, 192 / 16), 128, 0, stream>>>(
        wqkv, x, qkvpre, 192);
    dwconv_patchify<<<(192 * 65536) / 256, 256, 0, stream>>>(
        qkvpre, wdw, Qb[inp], Kb[inp], Vb[inp]);
    l2norm_rows<<<64 * 256, 256, 0, stream>>>(Qb[inp]);
    l2norm_rows<<<64 * 256, 256, 0, stream>>>(Kb[inp]);
  }

  // ---- Two crosses: (q1,k2,v1) and (q2,k1,v2)
  const dim3 gb(1, 16, 64);
  for (int cr = 0; cr < 2; ++cr) {
    const float* Qc = Qb[cr];
    const float* Kc = Kb[1 - cr];
    const float* Vc = Vb[cr];
    gemm_abt_batched<<<gb, 128, 0, stream>>>(Qc, Kc, attnXc, 1.0f);
    softmax_center<<<64 * 256, 256, 0, stream>>>(attnXc, temp);
    gemm_abt_batched<<<gb, 128, 0, stream>>>(attnXc, attnXc, covB,
                                             1.0f / 255.0f);
    jacobi_topk<<<64, 1024, JAC_SMEM_BYTES, stream>>>(covB, Vt, Ut);
    frmap_gemm<<<dim3(1, 7, 64), 128, 0, stream>>>(WfrP, Ut, Yt);
    proj_masked<<<gb, 128, 0, stream>>>(Yt, projM);
    gemm_ab_batched<<<gb, 128, 0, stream>>>(projM, Vc, Obuf);
    gather_shuffle4<<<(64 * 65536 / 4) / 256, 256, 0, stream>>>(Obuf, Gbuf);
    gemm_wn<64, 65536><<<dim3(65536 / 256, 64 / 16), 128, 0, stream>>>(
        wproj, Gbuf, out + (size_t)cr * CH, 64);
  }
}